// CMN_79774722556246
// MI455X (gfx1250) — compile-verified
//
#include <hip/hip_runtime.h>

typedef float v2f __attribute__((ext_vector_type(2)));
typedef float v8f __attribute__((ext_vector_type(8)));

#define N_PAIRS   131072
#define DIM       64
#define NS        10
#define PAIRS_PER_BLOCK 64
#define XP        68                    // padded LDS row in floats (bank-conflict-free)
#define PADF      (-4294967295.0f)      // -2^32 + 1

__global__ __launch_bounds__(128)
void cmn_kernel(const int* __restrict__ users,
                const int* __restrict__ items,
                const int* __restrict__ sampled_user,
                const float* __restrict__ emb_user,
                const float* __restrict__ emb_item,
                const float* __restrict__ W_w,
                const float* __restrict__ W_b,
                const float* __restrict__ U_w,
                const float* __restrict__ U_b,
                const float* __restrict__ bvec,
                const float* __restrict__ vvec,
                float* __restrict__ out)
{
    __shared__ float xs1[PAIRS_PER_BLOCK * XP];   // M*I  (GEMM A-operand for U_w)
    __shared__ float xs2[PAIRS_PER_BLOCK * XP];   // o    (GEMM A-operand for W_w)
    __shared__ float outp[4 * PAIRS_PER_BLOCK];   // per-wave partial row sums

    const int tid  = threadIdx.x;
    const int wave = tid >> 5;
    const int lane = tid & 31;
    const int p    = lane & 15;   // pair-in-tile (phase1) / matrix column (phase2)
    const int h    = lane >> 4;   // half-lane id

    // ================= Phase 1: gather + attention =================
    {
        const int pb = wave * 16 + p;                       // pair within block
        const int g  = blockIdx.x * PAIRS_PER_BLOCK + pb;   // global pair
        const int u  = users[g];
        const int it = items[g];

        const float4* Mp = (const float4*)(emb_user + (size_t)u  * DIM + 32 * h);
        const float4* Ip = (const float4*)(emb_item + (size_t)it * DIM + 32 * h);
        float4 sumv[8], prodv[8];
        #pragma unroll
        for (int i = 0; i < 8; ++i) {
            float4 m = Mp[i], iv = Ip[i];
            sumv[i]  = make_float4(m.x + iv.x, m.y + iv.y, m.z + iv.z, m.w + iv.w);
            prodv[i] = make_float4(m.x * iv.x, m.y * iv.y, m.z * iv.z, m.w * iv.w);
        }

        int aidx[NS];
        #pragma unroll
        for (int s = 0; s < NS; ++s) aidx[s] = sampled_user[(size_t)u * NS + s];

        // q_s = S * <M+I, adj_s>   (each lane does half the dot, combine across the pair)
        float q[NS];
        #pragma unroll
        for (int s = 0; s < NS; ++s) {
            const float4* Ap = (const float4*)(emb_user + (size_t)aidx[s] * DIM + 32 * h);
            float part = 0.0f;
            #pragma unroll
            for (int i = 0; i < 8; ++i) {
                float4 a = Ap[i];
                part = fmaf(sumv[i].x, a.x, part);
                part = fmaf(sumv[i].y, a.y, part);
                part = fmaf(sumv[i].z, a.z, part);
                part = fmaf(sumv[i].w, a.w, part);
            }
            float tot = part + __shfl_xor(part, 16, 32);
            float qs  = (float)NS * tot;
            q[s] = (qs != 0.0f) ? qs : PADF;
        }

        float qmax = q[0];
        #pragma unroll
        for (int s = 1; s < NS; ++s) qmax = fmaxf(qmax, q[s]);
        float es[NS], esum = 0.0f;
        #pragma unroll
        for (int s = 0; s < NS; ++s) { es[s] = __expf(q[s] - qmax); esum += es[s]; }
        const float inv = 1.0f / esum;

        // o = sum_s w_s * adj_s  (adj rows re-read: guaranteed L0/L2-hot)
        float4 ov[8];
        #pragma unroll
        for (int i = 0; i < 8; ++i) ov[i] = make_float4(0.f, 0.f, 0.f, 0.f);
        #pragma unroll
        for (int s = 0; s < NS; ++s) {
            const float4* Ap = (const float4*)(emb_user + (size_t)aidx[s] * DIM + 32 * h);
            const float w = es[s];
            #pragma unroll
            for (int i = 0; i < 8; ++i) {
                float4 a = Ap[i];
                ov[i].x = fmaf(w, a.x, ov[i].x);
                ov[i].y = fmaf(w, a.y, ov[i].y);
                ov[i].z = fmaf(w, a.z, ov[i].z);
                ov[i].w = fmaf(w, a.w, ov[i].w);
            }
        }

        float* r1 = xs1 + pb * XP + 32 * h;
        float* r2 = xs2 + pb * XP + 32 * h;
        #pragma unroll
        for (int i = 0; i < 8; ++i) {
            *(float4*)(r1 + 4 * i) = prodv[i];
            *(float4*)(r2 + 4 * i) = make_float4(ov[i].x * inv, ov[i].y * inv,
                                                 ov[i].z * inv, ov[i].w * inv);
        }
    }

    __syncthreads();

    // ================= Phase 2: pre = x1@U_w^T + x2@W_w^T via f32 WMMA =================
    // Wave `wave` owns output features d in [16*wave, 16*wave+16); iterates 4 pair tiles.
    {
        const int d = wave * 16 + p;        // this lane's output column (N of the C tile)

        // B fragments (4x16, K x N): VGPR j <-> K = j + 2h; element = W[d][4t+2h+j]
        v2f bU[16], bW[16];
        #pragma unroll
        for (int t = 0; t < 16; ++t) {
            bU[t] = *(const v2f*)(U_w + (size_t)d * DIM + 4 * t + 2 * h);
            bW[t] = *(const v2f*)(W_w + (size_t)d * DIM + 4 * t + 2 * h);
        }
        const float bias = U_b[d] + W_b[d] + bvec[d];
        const float vv   = vvec[d];

        for (int pt = 0; pt < 4; ++pt) {
            v8f acc;
            #pragma unroll
            for (int r = 0; r < 8; ++r) acc[r] = bias;   // bias broadcast into C

            // A fragment (16x4): lane row M = p, VGPR j <-> K = 4t + 2h + j
            const float* base1 = xs1 + (pt * 16 + p) * XP + 2 * h;
            const float* base2 = xs2 + (pt * 16 + p) * XP + 2 * h;
            #pragma unroll
            for (int t = 0; t < 16; ++t) {
                v2f a1 = *(const v2f*)(base1 + 4 * t);
                v2f a2 = *(const v2f*)(base2 + 4 * t);
                acc = __builtin_amdgcn_wmma_f32_16x16x4_f32(false, a1, false, bU[t],
                                                            (short)0, acc, false, false);
                acc = __builtin_amdgcn_wmma_f32_16x16x4_f32(false, a2, false, bW[t],
                                                            (short)0, acc, false, false);
            }

            // Epilogue: R = v[d] * leaky_relu(pre, 0.2); row-sum over the 16 d's of this wave.
            // C layout: VGPR r holds rows m=r (lanes 0-15) and m=r+8 (lanes 16-31).
            #pragma unroll
            for (int r = 0; r < 8; ++r) {
                float pre = acc[r];
                float lr  = (pre > 0.0f) ? pre : 0.2f * pre;
                float rv  = vv * lr;
                rv += __shfl_xor(rv, 1, 32);
                rv += __shfl_xor(rv, 2, 32);
                rv += __shfl_xor(rv, 4, 32);
                rv += __shfl_xor(rv, 8, 32);     // now all lanes in each half hold the row sum
                if (p == r)
                    outp[wave * PAIRS_PER_BLOCK + pt * 16 + r + 8 * h] = rv;
            }
        }
    }

    __syncthreads();

    // ================= Final cross-wave reduction over the 4 feature tiles =================
    if (tid < PAIRS_PER_BLOCK) {
        float s = outp[0 * PAIRS_PER_BLOCK + tid] + outp[1 * PAIRS_PER_BLOCK + tid]
                + outp[2 * PAIRS_PER_BLOCK + tid] + outp[3 * PAIRS_PER_BLOCK + tid];
        out[blockIdx.x * PAIRS_PER_BLOCK + tid] = s;
    }
}

extern "C" void kernel_launch(void* const* d_in, const int* in_sizes, int n_in,
                              void* d_out, int out_size, void* d_ws, size_t ws_size,
                              hipStream_t stream) {
    (void)in_sizes; (void)n_in; (void)d_ws; (void)ws_size; (void)out_size;
    const int*   users        = (const int*)  d_in[0];
    const int*   items        = (const int*)  d_in[1];
    const int*   sampled_user = (const int*)  d_in[2];
    const float* emb_user     = (const float*)d_in[3];
    const float* emb_item     = (const float*)d_in[4];
    const float* W_w          = (const float*)d_in[5];
    const float* W_b          = (const float*)d_in[6];
    const float* U_w          = (const float*)d_in[7];
    const float* U_b          = (const float*)d_in[8];
    const float* bvec         = (const float*)d_in[9];
    const float* vvec         = (const float*)d_in[10];
    float*       out          = (float*)d_out;

    dim3 grid(N_PAIRS / PAIRS_PER_BLOCK);   // 2048
    dim3 block(128);                        // 4 waves
    cmn_kernel<<<grid, block, 0, stream>>>(users, items, sampled_user, emb_user, emb_item,
                                           W_w, W_b, U_w, U_b, bvec, vvec, out);
}